// TransformerSequenceDecoder_75093208203778
// MI455X (gfx1250) — compile-verified
//
#include <hip/hip_runtime.h>
#include <hip/hip_bf16.h>
#include <math.h>

typedef float v2f __attribute__((ext_vector_type(2)));
typedef float v8f __attribute__((ext_vector_type(8)));

// ---- problem constants (from reference) ----
constexpr int cV      = 32000;
constexpr int cE      = 512;
constexpr int cNH     = 8;
constexpr int cNL     = 2;
constexpr int cFF     = 2048;
constexpr int cMAXLEN = 24;
constexpr int cS      = 48;
constexpr int cB      = 8;
constexpr int cINIT   = 12;                  // TRANS-2
constexpr int cLF     = cINIT + cMAXLEN - 1; // 35
constexpr int cTOK    = cLF * cB;            // 280

// GEMM staging: KC K-values per stage, padded LDS stride (16B-aligned, bank-spread)
constexpr int KC  = 16;
constexpr int KCP = KC + 4;                  // 20 dwords: n*20 mod 64 injective for n<16

// =====================================================================
// WMMA fp32 GEMM:  C[M,N] = A[M,K] @ W[N,K]^T + bias[N]  (optional relu)
// Block = 128 threads (4 waves) -> 16 rows x 256 cols.
// Double-buffered async global->LDS staging (ASYNCcnt) + ds_load fragments.
// Requires N % 256 == 0 and K % 16 == 0 (true for all uses here).
// =====================================================================
__global__ void __launch_bounds__(128)
k_wmma_gemm(const float* __restrict__ A, const float* __restrict__ W,
            const float* __restrict__ bias, float* __restrict__ C,
            int M, int N, int K, int ldc, int relu)
{
    __shared__ float Ws[2][256 * KCP];   // weight tile: 256 N-rows x KC K
    __shared__ float As[2][16  * KCP];   // activation tile: 16 M-rows x KC K

    const int tid  = threadIdx.x;
    const int lane = tid & 31;
    const int wave = tid >> 5;
    const int n0b  = blockIdx.x * 256;
    const int m0   = blockIdx.y * 16;

    const int half = lane >> 4;   // 0: lanes 0-15 hold K=k,k+1 ; 1: lanes 16-31 hold K=k+2,k+3
    const int l16  = lane & 15;
    const int koff = half * 2;

    // ---- async stage issue: 8 x b128 of W per thread (+1 of A for tid<64) ----
    auto issue_stage = [&](int pb, int kst) {
        const unsigned wbase = (unsigned)(size_t)(&Ws[pb][0]);
        #pragma unroll
        for (int i = 0; i < 8; ++i) {
            const int f = tid * 4 + i * 512;          // flat float idx in 256x16 tile
            const int r = f >> 4;                     // N-row 0..255
            const int c = f & 15;                     // K col (multiple of 4)
            const float* g = W + (size_t)(n0b + r) * K + kst + c;
            const unsigned l = wbase + (unsigned)((r * KCP + c) * 4);
            asm volatile("global_load_async_to_lds_b128 %0, %1, off"
                         :: "v"(l), "v"(g) : "memory");
        }
        if (tid < 64) {
            const unsigned abase = (unsigned)(size_t)(&As[pb][0]);
            const int f = tid * 4;                    // 16x16 tile = 256 floats
            const int r = f >> 4;
            const int c = f & 15;
            int row = m0 + r; if (row >= M) row = M - 1;   // clamp; masked on store
            const float* g = A + (size_t)row * K + kst + c;
            const unsigned l = abase + (unsigned)((r * KCP + c) * 4);
            asm volatile("global_load_async_to_lds_b128 %0, %1, off"
                         :: "v"(l), "v"(g) : "memory");
        }
    };

    v8f acc0 = {}, acc1 = {}, acc2 = {}, acc3 = {};
    const int S = K / KC;

    issue_stage(0, 0);
    for (int s = 0; s < S; ++s) {
        if (s + 1 < S) {
            issue_stage((s + 1) & 1, (s + 1) * KC);   // prefetch next stage
            // in-order completion: <=8 outstanding  =>  stage s fully landed
            asm volatile("s_wait_asynccnt 0x8" ::: "memory");
        } else {
            asm volatile("s_wait_asynccnt 0x0" ::: "memory");
        }
        __syncthreads();                               // stage s visible to all waves

        const float* as = &As[s & 1][l16 * KCP + koff];
        const float* w0 = &Ws[s & 1][(wave * 64 +  0 + l16) * KCP + koff];
        const float* w1 = w0 + 16 * KCP;
        const float* w2 = w0 + 32 * KCP;
        const float* w3 = w0 + 48 * KCP;
        #pragma unroll
        for (int kc = 0; kc < KC; kc += 4) {
            v2f a  = *(const v2f*)(as + kc);
            v2f b0 = *(const v2f*)(w0 + kc);
            v2f b1 = *(const v2f*)(w1 + kc);
            v2f b2 = *(const v2f*)(w2 + kc);
            v2f b3 = *(const v2f*)(w3 + kc);
            acc0 = __builtin_amdgcn_wmma_f32_16x16x4_f32(false, a, false, b0, (short)0, acc0, false, false);
            acc1 = __builtin_amdgcn_wmma_f32_16x16x4_f32(false, a, false, b1, (short)0, acc1, false, false);
            acc2 = __builtin_amdgcn_wmma_f32_16x16x4_f32(false, a, false, b2, (short)0, acc2, false, false);
            acc3 = __builtin_amdgcn_wmma_f32_16x16x4_f32(false, a, false, b3, (short)0, acc3, false, false);
        }
        __syncthreads();                               // buffer s&1 free for stage s+2
    }

    v8f accs[4] = {acc0, acc1, acc2, acc3};
    #pragma unroll
    for (int j = 0; j < 4; ++j) {
        const int   col = n0b + wave * 64 + j * 16 + l16;
        const float bv  = bias[col];
        #pragma unroll
        for (int r = 0; r < 8; ++r) {
            const int row = m0 + half * 8 + r;   // C/D layout: VGPR r, low lanes M=r, high M=8+r
            if (row < M) {
                float v = accs[j][r] + bv;
                if (relu) v = fmaxf(v, 0.0f);
                C[(size_t)row * ldc + col] = v;
            }
        }
    }
}

// =====================================================================
// Attention for one (b,h): scores -> softmax -> weighted sum of V.
// Layout: X[(t*B + b)*ld + h*64 + d]. blockDim = 64 (= head dim).
// =====================================================================
__global__ void __launch_bounds__(64)
k_attention(const float* __restrict__ Q, int ldq,
            const float* __restrict__ Km, int ldk,
            const float* __restrict__ Vm, int ldv,
            float* __restrict__ O, int ldo,
            int Lq, int Lk, int causal)
{
    __shared__ float Ks[cS * 64];
    __shared__ float Vs[cS * 64];
    __shared__ float qs[64];
    __shared__ float probs[64];
    __shared__ float red;

    const int t = threadIdx.x;
    const int b = blockIdx.x / cNH;
    const int h = blockIdx.x % cNH;
    const int hoff = h * 64;

    for (int k = 0; k < Lk; ++k) {
        Ks[k * 64 + t] = Km[(size_t)(k * cB + b) * ldk + hoff + t];
        Vs[k * 64 + t] = Vm[(size_t)(k * cB + b) * ldv + hoff + t];
    }
    __syncthreads();

    for (int q = 0; q < Lq; ++q) {
        qs[t] = Q[(size_t)(q * cB + b) * ldq + hoff + t];
        __syncthreads();

        float s = -1e30f;
        if (t < Lk) {
            float acc = 0.f;
            #pragma unroll 8
            for (int d = 0; d < 64; ++d) acc += qs[d] * Ks[t * 64 + d];
            s = acc * 0.125f;                  // 1/sqrt(64)
            if (causal && t > q) s += -1e9f;   // mask add, as in reference
        }
        probs[t] = s;
        __syncthreads();
        if (t == 0) {
            float m = -1e30f;
            for (int k = 0; k < Lk; ++k) m = fmaxf(m, probs[k]);
            red = m;
        }
        __syncthreads();
        float e = (t < Lk) ? expf(probs[t] - red) : 0.f;
        probs[t] = e;
        __syncthreads();
        if (t == 0) {
            float sum = 0.f;
            for (int k = 0; k < Lk; ++k) sum += probs[k];
            red = sum;
        }
        __syncthreads();
        const float inv = 1.0f / red;
        float o = 0.f;
        for (int k = 0; k < Lk; ++k) o += probs[k] * Vs[k * 64 + t];
        O[(size_t)(q * cB + b) * ldo + hoff + t] = o * inv;
        __syncthreads();
    }
}

// =====================================================================
// x = LayerNorm(x + y) * g + b ; one block (64 threads) per token, E=512.
// =====================================================================
__global__ void __launch_bounds__(64)
k_residual_ln(float* __restrict__ x, const float* __restrict__ y,
              const float* __restrict__ g, const float* __restrict__ b)
{
    __shared__ float red[64];
    const int t = threadIdx.x;
    const size_t base = (size_t)blockIdx.x * cE;

    float v[8];
    float s = 0.f;
    #pragma unroll
    for (int i = 0; i < 8; ++i) {
        const int e = t + i * 64;
        v[i] = x[base + e] + y[base + e];
        s += v[i];
    }
    red[t] = s; __syncthreads();
    for (int off = 32; off > 0; off >>= 1) { if (t < off) red[t] += red[t + off]; __syncthreads(); }
    const float mean = red[0] * (1.0f / cE);
    __syncthreads();

    float s2 = 0.f;
    #pragma unroll
    for (int i = 0; i < 8; ++i) { const float d = v[i] - mean; s2 += d * d; }
    red[t] = s2; __syncthreads();
    for (int off = 32; off > 0; off >>= 1) { if (t < off) red[t] += red[t + off]; __syncthreads(); }
    const float inv = rsqrtf(red[0] * (1.0f / cE) + 1e-5f);

    #pragma unroll
    for (int i = 0; i < 8; ++i) {
        const int e = t + i * 64;
        x[base + e] = (v[i] - mean) * inv * g[e] + b[e];
    }
}

// =====================================================================
// argmax over vocab for each batch row (first-max tie-break like jnp.argmax)
// =====================================================================
__global__ void __launch_bounds__(256)
k_argmax(const float* __restrict__ logits, int* __restrict__ pred)
{
    __shared__ float bv[256];
    __shared__ int   bi[256];
    const int t = threadIdx.x;
    const float* row = logits + (size_t)blockIdx.x * cV;

    float best = -1e30f; int besti = 0x7fffffff;
    for (int v = t; v < cV; v += 256) {
        const float f = row[v];
        if (f > best) { best = f; besti = v; }   // strided ascending -> first max kept
    }
    bv[t] = best; bi[t] = besti; __syncthreads();
    for (int off = 128; off > 0; off >>= 1) {
        if (t < off) {
            if (bv[t + off] > bv[t] || (bv[t + off] == bv[t] && bi[t + off] < bi[t])) {
                bv[t] = bv[t + off]; bi[t] = bi[t + off];
            }
        }
        __syncthreads();
    }
    if (t == 0) pred[blockIdx.x] = bi[0];
}

// ---- positional encoding table pe[MAXLEN][E] ----
__global__ void k_pe(float* __restrict__ pe)
{
    const int i = blockIdx.x * blockDim.x + threadIdx.x;
    if (i >= cMAXLEN * cE) return;
    const int l = i / cE, d = i % cE;
    const float div = expf(-logf(10000.0f) * (float)(d & ~1) / (float)cE);
    const float ang = (float)l * div;
    pe[i] = (d & 1) ? cosf(ang) : sinf(ang);
}

// ---- buf[:INIT] = emb[transform[1:-1]] + pe[:INIT]; rest zero ----
__global__ void k_embed_init(float* __restrict__ buf, const float* __restrict__ emb,
                             const int* __restrict__ transform, const float* __restrict__ pe)
{
    const int i = blockIdx.x * blockDim.x + threadIdx.x;
    if (i >= cTOK * cE) return;
    const int e  = i % cE;
    const int tb = i / cE;
    const int t  = tb / cB, b = tb % cB;
    float v = 0.f;
    if (t < cINIT) {
        const int tok = transform[(t + 1) * cB + b];
        v = emb[(size_t)tok * cE + e] + pe[t * cE + e];
    }
    buf[i] = v;
}

// ---- buf[pos+1][b] = emb[pred[b]] + pe[0] ----
__global__ void k_embed_pred(float* __restrict__ dst, const float* __restrict__ emb,
                             const int* __restrict__ pred, const float* __restrict__ pe0)
{
    const int i = blockIdx.x * blockDim.x + threadIdx.x;
    if (i >= cB * cE) return;
    const int b = i / cE, e = i % cE;
    dst[i] = emb[(size_t)pred[b] * cE + e] + pe0[e];
}

// =====================================================================
extern "C" void kernel_launch(void* const* d_in, const int* in_sizes, int n_in,
                              void* d_out, int out_size, void* d_ws, size_t ws_size,
                              hipStream_t stream)
{
    (void)in_sizes; (void)n_in; (void)out_size; (void)ws_size;

    const float* enc    = (const float*)d_in[0];
    const int*   transf = (const int*)  d_in[1];
    const float* emb    = (const float*)d_in[3];
    const float* sa_w   = (const float*)d_in[4];
    const float* sa_b   = (const float*)d_in[5];
    const float* sa_ow  = (const float*)d_in[6];
    const float* sa_ob  = (const float*)d_in[7];
    const float* ca_w   = (const float*)d_in[8];
    const float* ca_b   = (const float*)d_in[9];
    const float* ca_ow  = (const float*)d_in[10];
    const float* ca_ob  = (const float*)d_in[11];
    const float* ff1_w  = (const float*)d_in[12];
    const float* ff1_b  = (const float*)d_in[13];
    const float* ff2_w  = (const float*)d_in[14];
    const float* ff2_b  = (const float*)d_in[15];
    const float* ln_g   = (const float*)d_in[16];
    const float* ln_b   = (const float*)d_in[17];
    const float* out_w  = (const float*)d_in[18];
    const float* out_b  = (const float*)d_in[19];
    float* out = (float*)d_out;

    // workspace carve-up (~11 MB of floats)
    float* ws = (float*)d_ws;
    size_t o = 0;
    float* pe     = ws + o; o += (size_t)cMAXLEN * cE;
    float* buf    = ws + o; o += (size_t)cTOK * cE;      // pristine token buffer
    float* x      = ws + o; o += (size_t)cTOK * cE;      // activations
    float* y      = ws + o; o += (size_t)cTOK * cE;      // sublayer output
    float* qkv    = ws + o; o += (size_t)cTOK * 3 * cE;
    float* att    = ws + o; o += (size_t)cTOK * cE;
    float* qc     = ws + o; o += (size_t)cTOK * cE;
    float* ffh    = ws + o; o += (size_t)cTOK * cFF;
    float* kmem   = ws + o; o += (size_t)cNL * cS * cB * cE;
    float* vmem   = ws + o; o += (size_t)cNL * cS * cB * cE;
    float* logits = ws + o; o += (size_t)cB * cV;
    int*   pred   = (int*)(ws + o);

    auto gemm = [&](const float* A, const float* W, const float* bias, float* C,
                    int M, int N, int K, int ldc, int relu) {
        dim3 grid(N / 256, (M + 15) / 16);
        k_wmma_gemm<<<grid, 128, 0, stream>>>(A, W, bias, C, M, N, K, ldc, relu);
    };

    // one-time prep: PE table, initial embeddings, cross-attn K/V of memory
    k_pe<<<(cMAXLEN * cE + 255) / 256, 256, 0, stream>>>(pe);
    k_embed_init<<<(cTOK * cE + 255) / 256, 256, 0, stream>>>(buf, emb, transf, pe);
    for (int l = 0; l < cNL; ++l) {
        gemm(enc, ca_w + ((size_t)l * 3 * cE + cE)     * cE, ca_b + (size_t)l * 3 * cE + cE,
             kmem + (size_t)l * cS * cB * cE, cS * cB, cE, cE, cE, 0);
        gemm(enc, ca_w + ((size_t)l * 3 * cE + 2 * cE) * cE, ca_b + (size_t)l * 3 * cE + 2 * cE,
             vmem + (size_t)l * cS * cB * cE, cS * cB, cE, cE, cE, 0);
    }

    auto decoder_pass = [&]() {
        hipMemcpyAsync(x, buf, sizeof(float) * (size_t)cTOK * cE,
                       hipMemcpyDeviceToDevice, stream);
        for (int l = 0; l < cNL; ++l) {
            // --- self attention ---
            gemm(x, sa_w + (size_t)l * 3 * cE * cE, sa_b + (size_t)l * 3 * cE,
                 qkv, cTOK, 3 * cE, cE, 3 * cE, 0);
            k_attention<<<cB * cNH, 64, 0, stream>>>(qkv, 3 * cE, qkv + cE, 3 * cE,
                                                     qkv + 2 * cE, 3 * cE,
                                                     att, cE, cLF, cLF, 1);
            gemm(att, sa_ow + (size_t)l * cE * cE, sa_ob + (size_t)l * cE, y, cTOK, cE, cE, cE, 0);
            k_residual_ln<<<cTOK, 64, 0, stream>>>(x, y, ln_g + ((size_t)l * 3 + 0) * cE,
                                                          ln_b + ((size_t)l * 3 + 0) * cE);
            // --- cross attention (K/V precomputed) ---
            gemm(x, ca_w + (size_t)l * 3 * cE * cE, ca_b + (size_t)l * 3 * cE, qc, cTOK, cE, cE, cE, 0);
            k_attention<<<cB * cNH, 64, 0, stream>>>(qc, cE,
                                                     kmem + (size_t)l * cS * cB * cE, cE,
                                                     vmem + (size_t)l * cS * cB * cE, cE,
                                                     att, cE, cLF, cS, 0);
            gemm(att, ca_ow + (size_t)l * cE * cE, ca_ob + (size_t)l * cE, y, cTOK, cE, cE, cE, 0);
            k_residual_ln<<<cTOK, 64, 0, stream>>>(x, y, ln_g + ((size_t)l * 3 + 1) * cE,
                                                          ln_b + ((size_t)l * 3 + 1) * cE);
            // --- FFN ---
            gemm(x,   ff1_w + (size_t)l * cFF * cE, ff1_b + (size_t)l * cFF, ffh, cTOK, cFF, cE, cFF, 1);
            gemm(ffh, ff2_w + (size_t)l * cE * cFF, ff2_b + (size_t)l * cE,  y,   cTOK, cE, cFF, cE, 0);
            k_residual_ln<<<cTOK, 64, 0, stream>>>(x, y, ln_g + ((size_t)l * 3 + 2) * cE,
                                                          ln_b + ((size_t)l * 3 + 2) * cE);
        }
    };

    // autoregressive loop: 23 steps of (pass -> argmax -> embed)
    for (int i = 0; i < cMAXLEN - 1; ++i) {
        decoder_pass();
        const int pos = cINIT - 1 + i;
        gemm(x + (size_t)pos * cB * cE, out_w, out_b, logits, cB, cV, cE, cV, 0);
        k_argmax<<<cB, 256, 0, stream>>>(logits, pred);
        k_embed_pred<<<(cB * cE + 255) / 256, 256, 0, stream>>>(
            buf + (size_t)(pos + 1) * cB * cE, emb, pred, pe);
    }
    // final pass + full vocab projection -> d_out [LF, B, V]
    decoder_pass();
    gemm(x, out_w, out_b, out, cTOK, cV, cE, cV, 0);
}